// BiomarkerGNN_94489280770
// MI455X (gfx1250) — compile-verified
//
#include <hip/hip_runtime.h>
#include <hip/hip_bf16.h>
#include <math.h>

// ---------------------------------------------------------------------------
// BiomarkerGNN forward for gfx1250 (CDNA5).
// GEMMs: V_WMMA_F32_16X16X4_F32 (fp32 matrix path, wave32), A-tile streamed
// with GLOBAL_LOAD_ASYNC_TO_LDS_B128 (ASYNCcnt) and double-buffered in LDS;
// B (weight) fragments bulk-preloaded into registers per 32-K sub-block.
// ---------------------------------------------------------------------------

#define H 128
#define D_IN 256
#define BK 64
#define PAD 4   // row stride 68 floats = 272B -> 16B-aligned rows, no bank conflicts

typedef __attribute__((ext_vector_type(2))) float v2f;
typedef __attribute__((ext_vector_type(8))) float v8f;

__device__ __forceinline__ float gelu_exact(float v) {
    return 0.5f * v * (1.0f + erff(v * 0.70710678118654752f));
}

// CDNA5 async copy: copies 16 bytes from the global address g to the LDS address lds_off, tracked by ASYNCcnt.
__device__ __forceinline__ void async_ld_b128(unsigned lds_off, const float* g) {
    asm volatile("global_load_async_to_lds_b128 %0, %1, off"
                 :: "v"(lds_off), "v"(g) : "memory");
}
__device__ __forceinline__ void wait_async0() {
    asm volatile("s_wait_asynccnt 0x0" ::: "memory");
}
__device__ __forceinline__ unsigned lds_off_of(const void* p) {
    return (unsigned)(size_t)p;   // LDS aperture bits live in [63:32]
}

// ---------------------------------------------------------------------------
// WMMA GEMM: C[n x O] = A[n x K] @ W[O x K]^T + bias (+ residual) (opt. GELU)
// Block = 128 threads (4 waves). Block tile: 16 rows x 128 cols of C.
// Each wave: 16x32 tile = two v8f accumulators sharing one A fragment.
// A tile (16 x 64) double-buffered in LDS via async-to-LDS copies.
// Requires: n % 16 == 0, K % 64 == 0, O % 128 == 0.
// ---------------------------------------------------------------------------
__global__ __launch_bounds__(128) void gemm_wmma_kernel(
    const float* __restrict__ A, const float* __restrict__ W,
    const float* __restrict__ bias, const float* __restrict__ res,
    float* __restrict__ C, int n, int K, int O, int do_gelu)
{
    __shared__ float lds_a[2][16][BK + PAD];

    const int tid  = threadIdx.x;
    const int wave = tid >> 5;      // 0..3 -> 32-col group
    const int lane = tid & 31;
    const int half = lane >> 4;     // 0: lanes 0-15, 1: lanes 16-31
    const int l16  = lane & 15;

    const int row0 = blockIdx.x * 16;
    const int col0 = blockIdx.y * 128 + wave * 32;
    const int colA = col0 + l16;
    const int colB = col0 + 16 + l16;

    // cooperative A-tile mapping: thread t -> row t/8, 8 floats at col (t%8)*8
    const int la_r = tid >> 3;        // 0..15
    const int la_c = (tid & 7) * 8;   // 0..56

    // prefetch k-block 0 into buffer 0
    {
        const float* g = A + (long)(row0 + la_r) * K + la_c;
        async_ld_b128(lds_off_of(&lds_a[0][la_r][la_c]),     g);
        async_ld_b128(lds_off_of(&lds_a[0][la_r][la_c + 4]), g + 4);
    }
    wait_async0();
    __syncthreads();

    v8f acc0 = {};
    v8f acc1 = {};
    const int nkb = K / BK;

    for (int kb = 0; kb < nkb; ++kb) {
        const int cur = kb & 1;
        // issue async copy of next k-block into the other buffer
        if (kb + 1 < nkb) {
            const float* g = A + (long)(row0 + la_r) * K + (kb + 1) * BK + la_c;
            async_ld_b128(lds_off_of(&lds_a[cur ^ 1][la_r][la_c]),     g);
            async_ld_b128(lds_off_of(&lds_a[cur ^ 1][la_r][la_c + 4]), g + 4);
        }

        // B fragment base: B[k][col] = W[col][k]; lanes 0-15 -> K=kk,kk+1,
        // lanes 16-31 -> K=kk+2,kk+3 (2*half skew keeps 8B alignment)
        const float* wpa = W + (long)colA * K + kb * BK + 2 * half;
        const float* wpb = W + (long)colB * K + kb * BK + 2 * half;

        #pragma unroll
        for (int sub = 0; sub < 2; ++sub) {
            // bulk-preload 16 B fragments (one clause) for this 32-K sub-block
            v2f bA[8], bB[8];
            const float* wa = wpa + sub * 32;
            const float* wb = wpb + sub * 32;
            #pragma unroll
            for (int j = 0; j < 8; ++j) {
                bA[j] = *(const v2f*)(wa + 4 * j);
                bB[j] = *(const v2f*)(wb + 4 * j);
            }
            #pragma unroll
            for (int j = 0; j < 8; ++j) {
                const int kk = sub * 32 + j * 4;
                v2f a;
                a.x = lds_a[cur][l16][kk + 2 * half];
                a.y = lds_a[cur][l16][kk + 2 * half + 1];
                acc0 = __builtin_amdgcn_wmma_f32_16x16x4_f32(
                    false, a, false, bA[j], (short)0, acc0, false, false);
                acc1 = __builtin_amdgcn_wmma_f32_16x16x4_f32(
                    false, a, false, bB[j], (short)0, acc1, false, false);
            }
        }

        wait_async0();      // next tile fully landed in LDS
        __syncthreads();    // visible to all waves; safe to overwrite cur next iter
    }

    // D layout: VGPR r -> (M = r + 8*half, N = l16)
    const float bvA = bias ? bias[colA] : 0.0f;
    const float bvB = bias ? bias[colB] : 0.0f;
    #pragma unroll
    for (int r = 0; r < 8; ++r) {
        const int  row  = row0 + r + 8 * half;
        const long idxA = (long)row * O + colA;
        const long idxB = (long)row * O + colB;
        float va = acc0[r] + bvA;
        float vb = acc1[r] + bvB;
        if (res)     { va += res[idxA]; vb += res[idxB]; }
        if (do_gelu) { va = gelu_exact(va); vb = gelu_exact(vb); }
        C[idxA] = va;
        C[idxB] = vb;
    }
}

// ---------------------------------------------------------------------------
// Elementwise / reduction helpers
// ---------------------------------------------------------------------------
__global__ void fill_kernel(float* __restrict__ p, float v, long n) {
    long i = (long)blockIdx.x * blockDim.x + threadIdx.x;
    const long stride = (long)gridDim.x * blockDim.x;
    for (; i < n; i += stride) p[i] = v;
}

// column sums & sumsq; blockDim.x == C; sums/sumsq must be pre-zeroed
__global__ void colstats_kernel(const float* __restrict__ X, int n, int C,
                                float* __restrict__ sums, float* __restrict__ sumsq) {
    const int c = threadIdx.x;
    float s = 0.0f, s2 = 0.0f;
    for (int i = blockIdx.x; i < n; i += gridDim.x) {
        const float v = X[(long)i * C + c];
        s += v; s2 += v * v;
    }
    atomicAdd(&sums[c], s);
    atomicAdd(&sumsq[c], s2);
}

__global__ void bn_apply_kernel(const float* __restrict__ X, float* __restrict__ Y,
                                const float* __restrict__ sums, const float* __restrict__ sumsq,
                                const float* __restrict__ g, const float* __restrict__ b,
                                int n, int C, long total, int do_gelu) {
    long i = (long)blockIdx.x * blockDim.x + threadIdx.x;
    const long stride = (long)gridDim.x * blockDim.x;
    const float inv_n = 1.0f / (float)n;
    for (; i < total; i += stride) {
        const int c = (int)(i % C);
        const float mean = sums[c] * inv_n;
        const float var  = sumsq[c] * inv_n - mean * mean;
        float y = (X[i] - mean) * rsqrtf(var + 1e-5f) * g[c] + b[c];
        if (do_gelu) y = gelu_exact(y);
        Y[i] = y;
    }
}

// out = a + res + bias[c]
__global__ void add_bias_res_kernel(float* __restrict__ out, const float* __restrict__ a,
                                    const float* __restrict__ res, const float* __restrict__ bias,
                                    long total, int C) {
    long i = (long)blockIdx.x * blockDim.x + threadIdx.x;
    const long stride = (long)gridDim.x * blockDim.x;
    for (; i < total; i += stride) out[i] = a[i] + res[i] + bias[i % C];
}

// SAGE scatter: wave-per-edge; agg[dst] += h[src], cnt[dst] += 1
__global__ void sage_scatter_kernel(const float* __restrict__ h,
                                    const int* __restrict__ src, const int* __restrict__ dst,
                                    float* __restrict__ agg, float* __restrict__ cnt, int E) {
    const int gid  = blockIdx.x * blockDim.x + threadIdx.x;
    const int wid  = gid >> 5;
    const int lane = gid & 31;
    const int nw   = (gridDim.x * blockDim.x) >> 5;
    for (int e = wid; e < E; e += nw) {
        const int s = src[e], d = dst[e];
        const float* hs = h + (long)s * H;
        float* ad = agg + (long)d * H;
        #pragma unroll
        for (int c = lane; c < H; c += 32)
            atomicAdd(&ad[c], hs[c]);
        if (lane == 0) atomicAdd(&cnt[d], 1.0f);
    }
}

__global__ void deg_kernel(const int* __restrict__ dst, float* __restrict__ cnt, int E) {
    int i = blockIdx.x * blockDim.x + threadIdx.x;
    const int stride = gridDim.x * blockDim.x;
    for (; i < E; i += stride) atomicAdd(&cnt[dst[i]], 1.0f);
}

__global__ void dis_kernel(const float* __restrict__ deg, float* __restrict__ dis, int n) {
    int i = blockIdx.x * blockDim.x + threadIdx.x;
    const int stride = gridDim.x * blockDim.x;
    for (; i < n; i += stride) {
        const float d = deg[i];
        dis[i] = (d > 0.0f) ? rsqrtf(fmaxf(d, 1.0f)) : 0.0f;
    }
}

// GCN scatter: out[dst] += xw[src] * dis[src]*dis[dst]
__global__ void gcn_scatter_kernel(const float* __restrict__ xw,
                                   const int* __restrict__ src, const int* __restrict__ dst,
                                   const float* __restrict__ dis, float* __restrict__ out, int E) {
    const int gid  = blockIdx.x * blockDim.x + threadIdx.x;
    const int wid  = gid >> 5;
    const int lane = gid & 31;
    const int nw   = (gridDim.x * blockDim.x) >> 5;
    for (int e = wid; e < E; e += nw) {
        const int s = src[e], d = dst[e];
        const float nrm = dis[s] * dis[d];
        const float* xs = xw + (long)s * H;
        float* od = out + (long)d * H;
        #pragma unroll
        for (int c = lane; c < H; c += 32)
            atomicAdd(&od[c], xs[c] * nrm);
    }
}

// agg[i][c] /= max(cnt[i],1)
__global__ void meandiv_kernel(float* __restrict__ agg, const float* __restrict__ cnt, long total) {
    long i = (long)blockIdx.x * blockDim.x + threadIdx.x;
    const long stride = (long)gridDim.x * blockDim.x;
    for (; i < total; i += stride) {
        const int node = (int)(i / H);
        agg[i] = agg[i] / fmaxf(cnt[node], 1.0f);
    }
}

// out[i] = dot(h[i,:], Wp) + bp (O = 1 head)
__global__ void pred_kernel(const float* __restrict__ h, const float* __restrict__ Wp,
                            const float* __restrict__ bp, float* __restrict__ out, int n) {
    int i = blockIdx.x * blockDim.x + threadIdx.x;
    const int stride = gridDim.x * blockDim.x;
    for (; i < n; i += stride) {
        const float* hi = h + (long)i * H;
        float s = bp[0];
        #pragma unroll 8
        for (int c = 0; c < H; ++c) s += hi[c] * Wp[c];
        out[i] = s;
    }
}

// ---------------------------------------------------------------------------
// Host orchestration
// ---------------------------------------------------------------------------
static void run_gemm(const float* A, const float* W, const float* bias, const float* res,
                     float* C, int n, int K, int O, int do_gelu, hipStream_t st) {
    dim3 grid(n / 16, O / 128);
    gemm_wmma_kernel<<<grid, 128, 0, st>>>(A, W, bias, res, C, n, K, O, do_gelu);
}

static void run_fill(float* p, float v, long n, hipStream_t st) {
    int blocks = (int)((n + 255) / 256);
    if (blocks > 4096) blocks = 4096;
    fill_kernel<<<blocks, 256, 0, st>>>(p, v, n);
}

static void run_bn(const float* X, float* Y, const float* g, const float* b,
                   int n, int C, int do_gelu, float* stats, hipStream_t st) {
    run_fill(stats, 0.0f, 1024, st);
    colstats_kernel<<<256, C, 0, st>>>(X, n, C, stats, stats + 512);
    const long total = (long)n * C;
    bn_apply_kernel<<<4096, 256, 0, st>>>(X, Y, stats, stats + 512, g, b, n, C, total, do_gelu);
}

extern "C" void kernel_launch(void* const* d_in, const int* in_sizes, int n_in,
                              void* d_out, int out_size, void* d_ws, size_t ws_size,
                              hipStream_t stream) {
    const float* x  = (const float*)d_in[0];
    const int*   ei = (const int*)d_in[1];
    const int N_ = in_sizes[0] / D_IN;     // 50000
    const int E_ = in_sizes[1] / 2;        // 800000
    const int* src = ei;
    const int* dst = ei + E_;

    auto P = [&](int i) { return (const float*)d_in[i]; };

    // workspace carve-up (floats)
    float* ws = (float*)d_ws;
    long o = 0;
    float* bufA   = ws + o; o += (long)N_ * H;       // current h
    float* bufB   = ws + o; o += (long)N_ * H;       // skip / tmp
    float* bufC   = ws + o; o += (long)N_ * H;       // agg / xw / tmp
    float* bufD   = ws + o; o += (long)N_ * H;       // conv/ffn output
    float* bufBig = ws + o; o += (long)N_ * 4 * H;   // ffn hidden / post head
    float* cnt    = ws + o; o += N_;
    float* dis    = ws + o; o += N_;
    float* stats  = ws + o; o += 1024;
    if ((size_t)o * sizeof(float) > ws_size) return;  // insufficient scratch

    const long NH = (long)N_ * H;

    // ---- input projection: h = gelu(bn(x @ in_W.T + in_b))
    run_gemm(x, P(2), P(3), nullptr, bufB, N_, D_IN, H, 0, stream);
    run_bn(bufB, bufA, P(4), P(5), N_, H, /*gelu=*/1, stats, stream);

    int idx = 6;
    for (int i = 0; i < 4; ++i) {
        const float *Wl = nullptr, *bl = nullptr, *Wr = nullptr, *cW = nullptr, *cb = nullptr;
        if (i % 3 == 1) { cW = P(idx); cb = P(idx + 1); idx += 2; }
        else            { Wl = P(idx); bl = P(idx + 1); Wr = P(idx + 2); idx += 3; }
        const float* bn1g = P(idx);     const float* bn1b = P(idx + 1);
        const float* bn2g = P(idx + 2); const float* bn2b = P(idx + 3);
        const float* fW1  = P(idx + 4); const float* fb1  = P(idx + 5);
        const float* fW2  = P(idx + 6); const float* fb2  = P(idx + 7);
        const float* skW  = P(idx + 8); const float* skb  = P(idx + 9);
        idx += 10;

        // skip path
        const float* skip;
        if (i == 0) {
            skip = bufA;
        } else {
            run_gemm(bufA, skW, skb, nullptr, bufB, N_, H, H, 0, stream);
            skip = bufB;
        }

        if (i % 3 == 1) {
            // ---- GCNConv: out = segsum(xw[src]*dis[src]*dis[dst] -> dst) + b
            run_gemm(bufA, cW, nullptr, nullptr, bufC, N_, H, H, 0, stream);  // xw
            run_fill(cnt, 0.0f, N_, stream);
            deg_kernel<<<2048, 256, 0, stream>>>(dst, cnt, E_);
            dis_kernel<<<(N_ + 255) / 256, 256, 0, stream>>>(cnt, dis, N_);
            run_fill(bufD, 0.0f, NH, stream);
            gcn_scatter_kernel<<<4096, 256, 0, stream>>>(bufC, src, dst, dis, bufD, E_);
            // h_pre_bn = gcn_out + c_b + skip
            add_bias_res_kernel<<<4096, 256, 0, stream>>>(bufD, bufD, skip, cb, NH, H);
        } else {
            // ---- SAGEConv (mean): lin_l(mean nbr) + lin_r(x)
            run_fill(bufC, 0.0f, NH, stream);
            run_fill(cnt, 0.0f, N_, stream);
            sage_scatter_kernel<<<4096, 256, 0, stream>>>(bufA, src, dst, bufC, cnt, E_);
            meandiv_kernel<<<4096, 256, 0, stream>>>(bufC, cnt, NH);
            // bufD = mean @ Wl.T + bl + skip
            run_gemm(bufC, Wl, bl, skip, bufD, N_, H, H, 0, stream);
            // bufD += h @ Wr.T   (in-place residual)
            run_gemm(bufA, Wr, nullptr, bufD, bufD, N_, H, H, 0, stream);
        }

        // BN1 -> h1 (overwrites bufA; old h no longer needed)
        run_bn(bufD, bufA, bn1g, bn1b, N_, H, 0, stats, stream);

        // FFN: big = gelu(h1 @ W1.T + b1) ; out = big @ W2.T + b2 + h1
        run_gemm(bufA, fW1, fb1, nullptr, bufBig, N_, H, 4 * H, /*gelu=*/1, stream);
        run_gemm(bufBig, fW2, fb2, /*res=*/bufA, bufD, N_, 4 * H, H, 0, stream);

        // BN2 -> next h
        run_bn(bufD, bufA, bn2g, bn2b, N_, H, 0, stats, stream);
    }

    // ---- final BN + post head
    run_bn(bufA, bufB, P(57), P(58), N_, H, 0, stats, stream);                 // fin_bn
    run_gemm(bufB, P(59), P(60), nullptr, bufBig, N_, H, 2 * H, 0, stream);    // po1
    run_bn(bufBig, bufBig, P(61), P(62), N_, 2 * H, /*gelu=*/1, stats, stream);// po_bn1+gelu
    run_gemm(bufBig, P(63), P(64), nullptr, bufC, N_, 2 * H, H, 0, stream);    // po2
    run_bn(bufC, bufC, P(65), P(66), N_, H, /*gelu=*/1, stats, stream);        // po_bn2+gelu
    pred_kernel<<<(N_ + 255) / 256, 256, 0, stream>>>(bufC, P(67), P(68), (float*)d_out, N_);
}